// PredictionHead_88630945120561
// MI455X (gfx1250) — compile-verified
//
#include <hip/hip_runtime.h>
#include <math.h>

// ---------------------------------------------------------------------------
// Set2Set (N_ITERS=1) + MLP prediction head for MI455X (gfx1250, wave32).
// N_ITERS=1 with zero initial h/c/q_star makes the LSTM output a constant
// 64-vector q computed purely from (b_ih + b_hh). The heavy work is two
// streaming passes over feat (2 x 256MB, HBM bound ~22us), then a small MLP
// done with exact-fp32 V_WMMA_F32_16X16X4_F32 tiles fed from LDS (branch-free).
// ---------------------------------------------------------------------------

typedef float v2f __attribute__((ext_vector_type(2)));
typedef float v8f __attribute__((ext_vector_type(8)));

#define HDIM 64
#define ECAP 2048   // LDS cache for per-segment e values (avg seg len ~256)
#define APAD 132    // 128 + 4: keeps float4 LDS stores 16B-aligned, kills bank conflicts

__device__ __forceinline__ float eluf(float x) {
  return x > 0.0f ? x : (expf(x) - 1.0f);
}
__device__ __forceinline__ float sigf(float x) {
  return 1.0f / (1.0f + expf(-x));
}

// ---------------------------------------------------------------------------
// Kernel 1: q = LSTM step with zero state  ->  constant 64-vector from biases.
// gates layout (PyTorch): [i | f | g | o], each H wide. c0=0 kills the f gate.
// ---------------------------------------------------------------------------
__global__ void qvec_kernel(const float* __restrict__ b_ih,
                            const float* __restrict__ b_hh,
                            float* __restrict__ qvec) {
  int j = threadIdx.x;
  if (j >= HDIM) return;
  float ig = b_ih[0 * HDIM + j] + b_hh[0 * HDIM + j];
  float gg = b_ih[2 * HDIM + j] + b_hh[2 * HDIM + j];
  float og = b_ih[3 * HDIM + j] + b_hh[3 * HDIM + j];
  float c  = sigf(ig) * tanhf(gg);
  qvec[j]  = sigf(og) * tanhf(c);
}

// ---------------------------------------------------------------------------
// Kernel 2: segment offsets via binary search over sorted segment_ids.
// offs[b] = first index with seg[idx] >= b ; offs[B] = N. Empty segs OK.
// ---------------------------------------------------------------------------
__global__ void offsets_kernel(const int* __restrict__ seg,
                               int* __restrict__ offs, int N, int B) {
  int b = blockIdx.x * blockDim.x + threadIdx.x;
  if (b > B) return;
  int lo = 0, hi = N;
  while (lo < hi) {
    int mid = (lo + hi) >> 1;
    if (seg[mid] < b) lo = mid + 1; else hi = mid;
  }
  offs[b] = lo;
}

// ---------------------------------------------------------------------------
// Kernel 3: per-graph attention softmax + weighted readout (HBM-bound part).
// One block (8 waves) per graph. Wave-per-node: lane l loads feat[n][2l..2l+1]
// (float2 -> 256B coalesced per wave), butterfly-reduces the dot with q.
// Pass 1: segment max (e cached in LDS when len<=ECAP).
// Pass 2: denom = sum exp(e-max), readout += exp(e-max)*feat (LDS f32 atomics).
// Writes q_star row [q | readout/denom] to workspace.
// ---------------------------------------------------------------------------
__global__ void segment_kernel(const float* __restrict__ feat,
                               const int* __restrict__ offs,
                               const float* __restrict__ qvec,
                               float* __restrict__ qstar) {
  __shared__ float sQ[HDIM];
  __shared__ float sRead[HDIM];
  __shared__ float sWmax[8];
  __shared__ float sWden[8];
  __shared__ float sGmax;
  __shared__ float sDen;
  __shared__ float sE[ECAP];

  const int b     = blockIdx.x;
  const int tid   = threadIdx.x;
  const int lane  = tid & 31;
  const int wave  = tid >> 5;
  const int start = offs[b];
  const int len   = offs[b + 1] - start;
  const bool cacheE = (len <= ECAP);

  if (tid < HDIM) { sQ[tid] = qvec[tid]; sRead[tid] = 0.0f; }
  __syncthreads();

  const float qx = sQ[2 * lane];
  const float qy = sQ[2 * lane + 1];

  // ---- pass 1: max of e over segment ----
  float wmax = -INFINITY;
  for (int i = wave; i < len; i += 8) {
    const float2 f = ((const float2*)(feat + (size_t)(start + i) * HDIM))[lane];
    float v = f.x * qx + f.y * qy;
    #pragma unroll
    for (int s = 16; s >= 1; s >>= 1) v += __shfl_xor(v, s, 32);
    if (cacheE && lane == 0) sE[i] = v;
    wmax = fmaxf(wmax, v);
  }
  if (lane == 0) sWmax[wave] = wmax;
  __syncthreads();
  if (tid == 0) {
    float m = -INFINITY;
    #pragma unroll
    for (int w = 0; w < 8; ++w) m = fmaxf(m, sWmax[w]);
    sGmax = m;
  }
  __syncthreads();
  const float gmax = sGmax;

  // ---- pass 2: denom + weighted readout ----
  float rx = 0.0f, ry = 0.0f, den = 0.0f;
  for (int i = wave; i < len; i += 8) {
    const float2 f = ((const float2*)(feat + (size_t)(start + i) * HDIM))[lane];
    float e;
    if (cacheE) {
      e = sE[i];                       // written by this same wave in pass 1
    } else {
      float v = f.x * qx + f.y * qy;
      #pragma unroll
      for (int s = 16; s >= 1; s >>= 1) v += __shfl_xor(v, s, 32);
      e = v;
    }
    const float ex = expf(e - gmax);
    rx += ex * f.x;
    ry += ex * f.y;
    den += ex;                         // same value in all lanes; lane0 used
  }
  atomicAdd(&sRead[2 * lane],     rx);
  atomicAdd(&sRead[2 * lane + 1], ry);
  if (lane == 0) sWden[wave] = den;
  __syncthreads();
  if (tid == 0) {
    float d = 0.0f;
    #pragma unroll
    for (int w = 0; w < 8; ++w) d += sWden[w];
    sDen = d;
  }
  __syncthreads();

  if (tid < HDIM) {
    float r = (len > 0) ? (sRead[tid] / sDen) : 0.0f;  // empty seg -> 0 readout
    qstar[(size_t)b * 128 + tid]        = sQ[tid];
    qstar[(size_t)b * 128 + HDIM + tid] = r;
  }
}

// ---------------------------------------------------------------------------
// Kernel 4: MLP head with exact-fp32 WMMA.
//   x   = elu(q_star[B,128] @ lin0_w.T[128,64] + lin0_b)   via wmma 16x16x4
//   out = elu(x[B,64]       @ lin1_w.T[64,10]  + lin1_b)   scalar (tiny)
// Block = 128 threads (4 waves) handles 16 graph rows; wave w owns output
// columns [16w,16w+16). K=128 -> 32 wmma steps of K=4.
// The 16x128 q_star tile is staged into LDS once (guarded cooperative float4
// load), so the WMMA loop is completely branch-free: ds_load_b64 for A,
// global_load_b64 for B (L2-resident weights), EXEC all-1s throughout.
// A frag (16x4 f32): lanes 0-15 hold K={0,1}, lanes 16-31 K={2,3}, M=lane%16.
// C/D (16x16 f32): lane = N column (%16), vgpr r = row M (+8 for upper lanes).
// ---------------------------------------------------------------------------
__global__ void mlp_kernel(const float* __restrict__ qstar,
                           const float* __restrict__ lin0_w,
                           const float* __restrict__ lin0_b,
                           const float* __restrict__ lin1_w,
                           const float* __restrict__ lin1_b,
                           float* __restrict__ out, int B) {
  __shared__ float sA[16][APAD];   // staged q_star tile (padded rows)
  __shared__ float sX[16][HDIM];   // elu(lin0) activations

  const int tid  = threadIdx.x;
  const int lane = tid & 31;
  const int wave = tid >> 5;            // 0..3 -> output cols 16w..16w+15
  const int row0 = blockIdx.x * 16;
  const int m    = lane & 15;           // A-matrix row within tile
  const int hi   = lane >> 4;           // 0: K pair {0,1}, 1: K pair {2,3}
  const int ncol = wave * 16 + m;       // B/D column (0..63)

  // ---- stage 16x128 q_star tile into LDS (zero-pad OOB rows) ----
  for (int idx = tid; idx < 16 * 32; idx += 128) {     // 32 float4 per row
    const int r  = idx >> 5;
    const int cq = (idx & 31) * 4;
    const int row = row0 + r;
    float4 v = make_float4(0.0f, 0.0f, 0.0f, 0.0f);
    if (row < B) v = *(const float4*)(qstar + (size_t)row * 128 + cq);
    *(float4*)&sA[r][cq] = v;           // row stride 132f = 528B -> 16B aligned
  }
  __syncthreads();

  const float* brow = lin0_w + (size_t)ncol * 128;  // lin0_w is [64,128] row-major

  v8f acc = {};
  #pragma unroll
  for (int kk = 0; kk < 32; ++kk) {
    const int kb = kk * 4 + 2 * hi;
    const float2 af = *(const float2*)&sA[m][kb];   // 8B-aligned ds_load_b64
    const float2 bf = *(const float2*)(brow + kb);  // 8B-aligned global b64
    v2f a, bm;
    a.x = af.x;  a.y = af.y;
    bm.x = bf.x; bm.y = bf.y;
    acc = __builtin_amdgcn_wmma_f32_16x16x4_f32(
        /*neg_a=*/false, a, /*neg_b=*/false, bm,
        /*c_mod=*/(short)0, acc, /*reuse_a=*/false, /*reuse_b=*/false);
  }

  const float bias0 = lin0_b[ncol];
  #pragma unroll
  for (int r = 0; r < 8; ++r) {
    const int mrow = r + 8 * hi;        // D row within the 16-row tile
    sX[mrow][ncol] = eluf(acc[r] + bias0);
  }
  __syncthreads();

  // lin1: 16 rows x 10 outputs = 160 dot products over 64 (from LDS)
  for (int idx = tid; idx < 16 * 10; idx += blockDim.x) {
    const int mm = idx / 10;
    const int oo = idx % 10;
    if (row0 + mm >= B) continue;
    float s = lin1_b[oo];
    const float* wrow = lin1_w + oo * HDIM;
    #pragma unroll
    for (int n = 0; n < HDIM; ++n) s += sX[mm][n] * wrow[n];
    out[(size_t)(row0 + mm) * 10 + oo] = eluf(s);
  }
}

// ---------------------------------------------------------------------------
// Launch. Inputs (setup_inputs order):
//  0 feat[N,64] 1 w_ih 2 w_hh 3 b_ih 4 b_hh 5 lin0_w[64,128] 6 lin0_b[64]
//  7 lin1_w[10,64] 8 lin1_b[10] 9 segment_ids[N] 10 num_graphs(=4096)
// w_ih/w_hh are mathematically dead for N_ITERS=1 (multiplied by zero state).
// ---------------------------------------------------------------------------
extern "C" void kernel_launch(void* const* d_in, const int* in_sizes, int n_in,
                              void* d_out, int out_size, void* d_ws, size_t ws_size,
                              hipStream_t stream) {
  const float* feat   = (const float*)d_in[0];
  const float* b_ih   = (const float*)d_in[3];
  const float* b_hh   = (const float*)d_in[4];
  const float* lin0_w = (const float*)d_in[5];
  const float* lin0_b = (const float*)d_in[6];
  const float* lin1_w = (const float*)d_in[7];
  const float* lin1_b = (const float*)d_in[8];
  const int*   seg    = (const int*)d_in[9];
  float* out = (float*)d_out;

  const int N = in_sizes[9];        // node count
  const int B = out_size / 10;      // graph count (output is [B,10])

  // workspace layout: [offs: (B+1) ints][qvec: 64 f][qstar: B*128 f]
  char* ws = (char*)d_ws;
  size_t o1 = ((size_t)(B + 1) * sizeof(int) + 255) & ~(size_t)255;
  int*   offs  = (int*)ws;
  float* qvec  = (float*)(ws + o1);
  float* qstar = (float*)(ws + o1 + 256);

  qvec_kernel<<<1, HDIM, 0, stream>>>(b_ih, b_hh, qvec);
  offsets_kernel<<<(B + 1 + 255) / 256, 256, 0, stream>>>(seg, offs, N, B);
  segment_kernel<<<B, 256, 0, stream>>>(feat, offs, qvec, qstar);
  mlp_kernel<<<(B + 15) / 16, 128, 0, stream>>>(qstar, lin0_w, lin0_b,
                                                lin1_w, lin1_b, out, B);
}